// HGT_66400194396401
// MI455X (gfx1250) — compile-verified
//
#include <hip/hip_runtime.h>
#include <math.h>

// ---------------------------------------------------------------------------
// HGT forward for MI455X (gfx1250, wave32). All GEMMs via V_WMMA_F32_16X16X4_F32
// (full f32 precision matching the reference), B operand staged through LDS.
// Edge softmax via L2 atomics.
// ---------------------------------------------------------------------------

typedef __attribute__((ext_vector_type(2))) float v2f;
typedef __attribute__((ext_vector_type(8))) float v8f;

#define HID   512
#define NHEAD 8
#define HD    64
#define NGENE 40000
#define NTRAIT 10000
#define OUTC  128

__device__ inline float gelu_f(float x) {
  return 0.5f * x * (1.0f + erff(x * 0.70710678118654752f));
}

// order-preserving float atomic max (init with -inf)
__device__ inline void atomicMaxF(float* addr, float val) {
  if (!(val == val)) return;
  int iv = __float_as_int(val);
  if (iv >= 0) atomicMax((int*)addr, iv);
  else         atomicMin((unsigned int*)addr, (unsigned int)iv);
}

// ---------------------------------------------------------------------------
// Batched f32 WMMA GEMM:  C[z] = A[z] @ B[z] (+ bias)
// Block = 256 threads = 8 waves. Block tile: 128 (M) x 64 (N).
// Each wave owns a 16x64 C strip (4 v8f accumulators).
// B is staged through LDS in 32-row K-chunks (8 KB), shared by all 8 waves;
// columns are swizzled by ((row&3)<<4) to keep the two half-waves (which read
// K-rows differing by 2) on disjoint LDS banks.
// WMMA fragments (V_WMMA_F32_16X16X4_F32):
//   A 16x4: lanes 0-15 hold K={0,1}, lanes 16-31 hold K={2,3} (one b64 load)
//   B 4x16: lanes 0-15 = N, v0=row kk, v1=row kk+1
//   C 16x16: 8 VGPRs, M-major, lanes 16-31 offset M by 8
// Requires: M % 16 == 0, N % 64 == 0, K % 32 == 0.
// ---------------------------------------------------------------------------
__global__ __launch_bounds__(256) void gemm_wmma_f32(
    const float* __restrict__ A, const float* __restrict__ B,
    const float* __restrict__ bias, float* __restrict__ C,
    int M, int N, int K, int lda, int ldb, int ldc,
    long long aBatch, long long bBatch, long long cBatch)
{
  __shared__ float Bs[32 * 64];

  int wave = threadIdx.x >> 5;
  int lane = threadIdx.x & 31;
  int row0 = (blockIdx.x * 8 + wave) * 16;
  int col0 = blockIdx.y * 64;
  bool valid = (row0 < M);
  int rowc = valid ? row0 : (M - 16);   // clamp so OOB waves still load legal addrs

  const float* Ab = A + (long long)blockIdx.z * aBatch;
  const float* Bb = B + (long long)blockIdx.z * bBatch;
  float*       Cb = C + (long long)blockIdx.z * cBatch;

  int m  = lane & 15;
  int kk = (lane >> 4) << 1;   // 0 or 2
  int n  = lane & 15;

  // staging coordinates (per thread): row r of the 32-row chunk, 8 columns
  int sr = threadIdx.x >> 3;         // 0..31
  int sc = (threadIdx.x & 7) << 3;   // 0,8,...,56
  int scs = (sc + ((sr & 3) << 4)) & 63;  // swizzled column base (multiple of 8)

  v8f acc[4] = {};
  for (int k0 = 0; k0 < K; k0 += 32) {
    __syncthreads();   // previous chunk fully consumed
    {
      const float* bg = Bb + (long long)(k0 + sr) * ldb + col0 + sc;
      float4 b0 = *(const float4*)bg;
      float4 b1 = *(const float4*)(bg + 4);
      *(float4*)&Bs[sr * 64 + scs]     = b0;
      *(float4*)&Bs[sr * 64 + scs + 4] = b1;
    }
    __syncthreads();   // chunk visible to all waves

#pragma unroll
    for (int ks = 0; ks < 8; ++ks) {
      const float* ap = Ab + (long long)(rowc + m) * lda + (k0 + ks * 4 + kk);
      v2f a;
      a.x = ap[0];
      a.y = ap[1];
      int r0 = ks * 4 + kk;      // B row for b.x (b.y = r0 + 1)
#pragma unroll
      for (int j = 0; j < 4; ++j) {
        int jc = j * 16 + n;
        v2f b;
        b.x = Bs[r0 * 64       + ((jc + (( r0      & 3) << 4)) & 63)];
        b.y = Bs[(r0 + 1) * 64 + ((jc + (((r0 + 1) & 3) << 4)) & 63)];
        acc[j] = __builtin_amdgcn_wmma_f32_16x16x4_f32(
            false, a, false, b, (short)0, acc[j], false, false);
      }
    }
  }

  if (!valid) return;
  int moff = (lane >> 4) << 3;  // 0 or 8
#pragma unroll
  for (int j = 0; j < 4; ++j) {
    int cc = col0 + j * 16 + n;
    float bv = bias ? bias[cc] : 0.0f;
#pragma unroll
    for (int r = 0; r < 8; ++r) {
      Cb[(long long)(row0 + moff + r) * ldc + cc] = acc[j][r] + bv;
    }
  }
}

__global__ void fill_f32(float* __restrict__ p, float v, long long n) {
  long long i = (long long)blockIdx.x * blockDim.x + threadIdx.x;
  if (i < n) p[i] = v;
}

__global__ void gelu_inplace(float* __restrict__ x, long long n) {
  long long i = (long long)blockIdx.x * blockDim.x + threadIdx.x;
  if (i < n) x[i] = gelu_f(x[i]);
}

// alpha[e,h] = dot(q[dst,h,:], Ks[src,h,:]) * p[h] / sqrt(D); segment max via atomics
__global__ __launch_bounds__(256) void edge_alpha(
    const int* __restrict__ src, const int* __restrict__ dst, int E,
    const float* __restrict__ Ks,    // [Ns, HID]
    const float* __restrict__ KQVd,  // [Nd, 3*HID], q at col offset HID
    const float* __restrict__ prel,  // [NHEAD]
    float* __restrict__ alpha,       // [E, NHEAD]
    float* __restrict__ mmax)        // [Nd, NHEAD]
{
  int t = blockIdx.x * blockDim.x + threadIdx.x;
  if (t >= E * NHEAD) return;
  int e = t >> 3, h = t & 7;
  int s = src[e], d = dst[e];
  const float4* kp = (const float4*)(Ks   + (long long)s * HID + h * HD);
  const float4* qp = (const float4*)(KQVd + (long long)d * (3 * HID) + HID + h * HD);
  float sum = 0.0f;
#pragma unroll
  for (int i = 0; i < HD / 4; ++i) {
    float4 kv = kp[i], qv = qp[i];
    sum += kv.x * qv.x + kv.y * qv.y + kv.z * qv.z + kv.w * qv.w;
  }
  float av = sum * prel[h] * 0.125f;  // 1/sqrt(64)
  alpha[t] = av;
  atomicMaxF(&mmax[(long long)d * NHEAD + h], av);
}

// w = exp(alpha - m[dst]); alpha <- w; denom[dst] += w
__global__ __launch_bounds__(256) void edge_expsum(
    const int* __restrict__ dst, int E,
    const float* __restrict__ mmax, float* __restrict__ alpha,
    float* __restrict__ den)
{
  int t = blockIdx.x * blockDim.x + threadIdx.x;
  if (t >= E * NHEAD) return;
  int e = t >> 3, h = t & 7;
  int d = dst[e];
  float w = expf(alpha[t] - mmax[(long long)d * NHEAD + h]);
  alpha[t] = w;
  atomicAdd(&den[(long long)d * NHEAD + h], w);
}

// agg[dst,h,:] += (w/denom[dst,h]) * Vs[src,h,:]
__global__ __launch_bounds__(256) void edge_agg(
    const int* __restrict__ src, const int* __restrict__ dst, int E,
    const float* __restrict__ alpha, const float* __restrict__ den,
    const float* __restrict__ Vs, float* __restrict__ agg)
{
  int t = blockIdx.x * blockDim.x + threadIdx.x;
  if (t >= E * NHEAD) return;
  int e = t >> 3, h = t & 7;
  int s = src[e], d = dst[e];
  float coef = alpha[t] / den[(long long)d * NHEAD + h];
  const float4* vp = (const float4*)(Vs + (long long)s * HID + h * HD);
  float* ap = agg + (long long)d * HID + h * HD;
#pragma unroll
  for (int i = 0; i < HD / 4; ++i) {
    float4 v = vp[i];
    atomicAdd(ap + i * 4 + 0, coef * v.x);
    atomicAdd(ap + i * 4 + 1, coef * v.y);
    atomicAdd(ap + i * 4 + 2, coef * v.z);
    atomicAdd(ap + i * 4 + 3, coef * v.w);
  }
}

// x <- gelu( sigmoid(skip)*o + (1-sigmoid(skip))*x )
__global__ void blend_gelu(const float* __restrict__ o, float* __restrict__ x,
                           const float* __restrict__ skip, long long n)
{
  long long i = (long long)blockIdx.x * blockDim.x + threadIdx.x;
  if (i >= n) return;
  float a = 1.0f / (1.0f + expf(-skip[0]));
  float v = a * o[i] + (1.0f - a) * x[i];
  x[i] = gelu_f(v);
}

static inline int cdiv(long long a, long long b) { return (int)((a + b - 1) / b); }

extern "C" void kernel_launch(void* const* d_in, const int* in_sizes, int n_in,
                              void* d_out, int out_size, void* d_ws, size_t ws_size,
                              hipStream_t stream)
{
  (void)in_sizes; (void)n_in; (void)out_size; (void)ws_size;
  // -------- inputs (setup_inputs dict order; params flattened as jax pytree:
  // keys sorted -> Wproj, bproj, layers[0..1]; per layer:
  // Ak{gg,gt,tg}, Av{gg,gt,tg}, Wkqv{g,t}, Wout{g,t}, bkqv{g,t}, bout{g,t},
  // p{gg,gt,tg}, skip{g,t} ) --------
  const float* x_gene  = (const float*)d_in[0];
  const float* x_trait = (const float*)d_in[1];
  const int* ei_gt = (const int*)d_in[2];   // [2,100000]: src | dst
  const int* ei_tg = (const int*)d_in[3];
  const int* ei_gg = (const int*)d_in[4];
  const float* Wproj = (const float*)d_in[5];
  const float* bproj = (const float*)d_in[6];
  const int Egt = 100000, Etg = 100000, Egg = 200000;

  // -------- workspace layout (floats) --------
  float* ws = (float*)d_ws;
  size_t off = 0;
  auto alloc = [&](size_t nel) { float* p = ws + off; off += nel; return p; };
  float* XG    = alloc((size_t)NGENE * HID);
  float* XT    = alloc((size_t)NTRAIT * HID);
  float* KQVG  = alloc((size_t)NGENE * 3 * HID);
  float* KQVT  = alloc((size_t)NTRAIT * 3 * HID);
  float* KS_GT = alloc((size_t)NGENE * HID);
  float* VS_GT = alloc((size_t)NGENE * HID);
  float* KS_TG = alloc((size_t)NTRAIT * HID);
  float* VS_TG = alloc((size_t)NTRAIT * HID);
  float* KS_GG = alloc((size_t)NGENE * HID);
  float* VS_GG = alloc((size_t)NGENE * HID);
  float* AL_GT = alloc((size_t)Egt * NHEAD);
  float* AL_TG = alloc((size_t)Etg * NHEAD);
  float* AL_GG = alloc((size_t)Egg * NHEAD);
  float* MG   = alloc((size_t)NGENE * NHEAD);
  float* MT   = alloc((size_t)NTRAIT * NHEAD);
  float* DENG = alloc((size_t)NGENE * NHEAD);
  float* DENT = alloc((size_t)NTRAIT * NHEAD);
  float* AGGG = alloc((size_t)NGENE * HID);
  float* AGGT = alloc((size_t)NTRAIT * HID);
  float* OG   = alloc((size_t)NGENE * HID);
  float* OT   = alloc((size_t)NTRAIT * HID);

  hipMemcpyAsync(XG, x_gene,  (size_t)NGENE  * HID * sizeof(float), hipMemcpyDeviceToDevice, stream);
  hipMemcpyAsync(XT, x_trait, (size_t)NTRAIT * HID * sizeof(float), hipMemcpyDeviceToDevice, stream);

  dim3 blk(256, 1, 1);
  for (int lp = 0; lp < 2; ++lp) {
    int b = 7 + lp * 19;
    const float* Ak_gg = (const float*)d_in[b + 0];
    const float* Ak_gt = (const float*)d_in[b + 1];
    const float* Ak_tg = (const float*)d_in[b + 2];
    const float* Av_gg = (const float*)d_in[b + 3];
    const float* Av_gt = (const float*)d_in[b + 4];
    const float* Av_tg = (const float*)d_in[b + 5];
    const float* Wkqv_g = (const float*)d_in[b + 6];
    const float* Wkqv_t = (const float*)d_in[b + 7];
    const float* Wout_g = (const float*)d_in[b + 8];
    const float* Wout_t = (const float*)d_in[b + 9];
    const float* bkqv_g = (const float*)d_in[b + 10];
    const float* bkqv_t = (const float*)d_in[b + 11];
    const float* bout_g = (const float*)d_in[b + 12];
    const float* bout_t = (const float*)d_in[b + 13];
    const float* p_gg = (const float*)d_in[b + 14];
    const float* p_gt = (const float*)d_in[b + 15];
    const float* p_tg = (const float*)d_in[b + 16];
    const float* skip_g = (const float*)d_in[b + 17];
    const float* skip_t = (const float*)d_in[b + 18];

    // --- fused KQV projections: [N,512] @ [512,1536] + b ---
    {
      dim3 g(cdiv(NGENE, 128), 1536 / 64, 1);
      gemm_wmma_f32<<<g, blk, 0, stream>>>(XG, Wkqv_g, bkqv_g, KQVG,
          NGENE, 1536, HID, HID, 1536, 1536, 0, 0, 0);
      dim3 gt(cdiv(NTRAIT, 128), 1536 / 64, 1);
      gemm_wmma_f32<<<gt, blk, 0, stream>>>(XT, Wkqv_t, bkqv_t, KQVT,
          NTRAIT, 1536, HID, HID, 1536, 1536, 0, 0, 0);
    }

    // --- per-head relation transforms: batched over 8 heads (z) ---
    {
      dim3 gg(cdiv(NGENE, 128), 1, NHEAD);
      dim3 gt(cdiv(NTRAIT, 128), 1, NHEAD);
      // relation gt (src gene)
      gemm_wmma_f32<<<gg, blk, 0, stream>>>(KQVG,            Ak_gt, nullptr, KS_GT,
          NGENE, HD, HD, 3 * HID, HD, HID, HD, HD * HD, HD);
      gemm_wmma_f32<<<gg, blk, 0, stream>>>(KQVG + 2 * HID,  Av_gt, nullptr, VS_GT,
          NGENE, HD, HD, 3 * HID, HD, HID, HD, HD * HD, HD);
      // relation tg (src trait)
      gemm_wmma_f32<<<gt, blk, 0, stream>>>(KQVT,            Ak_tg, nullptr, KS_TG,
          NTRAIT, HD, HD, 3 * HID, HD, HID, HD, HD * HD, HD);
      gemm_wmma_f32<<<gt, blk, 0, stream>>>(KQVT + 2 * HID,  Av_tg, nullptr, VS_TG,
          NTRAIT, HD, HD, 3 * HID, HD, HID, HD, HD * HD, HD);
      // relation gg (src gene)
      gemm_wmma_f32<<<gg, blk, 0, stream>>>(KQVG,            Ak_gg, nullptr, KS_GG,
          NGENE, HD, HD, 3 * HID, HD, HID, HD, HD * HD, HD);
      gemm_wmma_f32<<<gg, blk, 0, stream>>>(KQVG + 2 * HID,  Av_gg, nullptr, VS_GG,
          NGENE, HD, HD, 3 * HID, HD, HID, HD, HD * HD, HD);
    }

    // --- init segment-softmax state ---
    fill_f32<<<cdiv((long long)NGENE * NHEAD, 256), 256, 0, stream>>>(MG, -INFINITY, (long long)NGENE * NHEAD);
    fill_f32<<<cdiv((long long)NTRAIT * NHEAD, 256), 256, 0, stream>>>(MT, -INFINITY, (long long)NTRAIT * NHEAD);
    fill_f32<<<cdiv((long long)NGENE * NHEAD, 256), 256, 0, stream>>>(DENG, 0.0f, (long long)NGENE * NHEAD);
    fill_f32<<<cdiv((long long)NTRAIT * NHEAD, 256), 256, 0, stream>>>(DENT, 0.0f, (long long)NTRAIT * NHEAD);
    fill_f32<<<cdiv((long long)NGENE * HID, 256), 256, 0, stream>>>(AGGG, 0.0f, (long long)NGENE * HID);
    fill_f32<<<cdiv((long long)NTRAIT * HID, 256), 256, 0, stream>>>(AGGT, 0.0f, (long long)NTRAIT * HID);

    // --- edge attention logits + segment max ---
    edge_alpha<<<cdiv((long long)Egt * NHEAD, 256), 256, 0, stream>>>(
        ei_gt, ei_gt + Egt, Egt, KS_GT, KQVT, p_gt, AL_GT, MT);
    edge_alpha<<<cdiv((long long)Etg * NHEAD, 256), 256, 0, stream>>>(
        ei_tg, ei_tg + Etg, Etg, KS_TG, KQVG, p_tg, AL_TG, MG);
    edge_alpha<<<cdiv((long long)Egg * NHEAD, 256), 256, 0, stream>>>(
        ei_gg, ei_gg + Egg, Egg, KS_GG, KQVG, p_gg, AL_GG, MG);

    // --- exp + segment sum ---
    edge_expsum<<<cdiv((long long)Egt * NHEAD, 256), 256, 0, stream>>>(ei_gt + Egt, Egt, MT, AL_GT, DENT);
    edge_expsum<<<cdiv((long long)Etg * NHEAD, 256), 256, 0, stream>>>(ei_tg + Etg, Etg, MG, AL_TG, DENG);
    edge_expsum<<<cdiv((long long)Egg * NHEAD, 256), 256, 0, stream>>>(ei_gg + Egg, Egg, MG, AL_GG, DENG);

    // --- weighted aggregation ---
    edge_agg<<<cdiv((long long)Egt * NHEAD, 256), 256, 0, stream>>>(
        ei_gt, ei_gt + Egt, Egt, AL_GT, DENT, VS_GT, AGGT);
    edge_agg<<<cdiv((long long)Etg * NHEAD, 256), 256, 0, stream>>>(
        ei_tg, ei_tg + Etg, Etg, AL_TG, DENG, VS_TG, AGGG);
    edge_agg<<<cdiv((long long)Egg * NHEAD, 256), 256, 0, stream>>>(
        ei_gg, ei_gg + Egg, Egg, AL_GG, DENG, VS_GG, AGGG);

    // --- agg <- gelu(agg), then O = agg @ Wout + bout ---
    gelu_inplace<<<cdiv((long long)NGENE * HID, 256), 256, 0, stream>>>(AGGG, (long long)NGENE * HID);
    gelu_inplace<<<cdiv((long long)NTRAIT * HID, 256), 256, 0, stream>>>(AGGT, (long long)NTRAIT * HID);
    {
      dim3 g(cdiv(NGENE, 128), HID / 64, 1);
      gemm_wmma_f32<<<g, blk, 0, stream>>>(AGGG, Wout_g, bout_g, OG,
          NGENE, HID, HID, HID, HID, HID, 0, 0, 0);
      dim3 gt(cdiv(NTRAIT, 128), HID / 64, 1);
      gemm_wmma_f32<<<gt, blk, 0, stream>>>(AGGT, Wout_t, bout_t, OT,
          NTRAIT, HID, HID, HID, HID, HID, 0, 0, 0);
    }

    // --- skip blend + outer GELU, in place on running x ---
    blend_gelu<<<cdiv((long long)NGENE * HID, 256), 256, 0, stream>>>(OG, XG, skip_g, (long long)NGENE * HID);
    blend_gelu<<<cdiv((long long)NTRAIT * HID, 256), 256, 0, stream>>>(OT, XT, skip_t, (long long)NTRAIT * HID);
  }

  // --- final gene projection: [40000,512] @ [512,128] + bproj -> d_out ---
  {
    dim3 g(cdiv(NGENE, 128), OUTC / 64, 1);
    gemm_wmma_f32<<<g, blk, 0, stream>>>(XG, Wproj, bproj, (float*)d_out,
        NGENE, OUTC, HID, HID, OUTC, OUTC, 0, 0, 0);
  }
  // --- trait output: x['trait'] after layer 2 ---
  hipMemcpyAsync((float*)d_out + (size_t)NGENE * OUTC, XT,
                 (size_t)NTRAIT * HID * sizeof(float), hipMemcpyDeviceToDevice, stream);
}